// PraxisMemory_11562051961024
// MI455X (gfx1250) — compile-verified
//
#include <hip/hip_runtime.h>
#include <hip/hip_bf16.h>

// ---------------------------------------------------------------------------
// PraxisMemory fused KNN-attention for MI455X (gfx1250, wave32).
//   sims GEMM via V_WMMA_F32_16X16X4_F32, fused with per-query top-16
//   selection, value gather, weighted sum and sigmoid-gate blend.
// Fixed reference dims: B=1, H=16, D=128, TOP_K=16 (S, M derived at launch).
//
// v2: A tile pre-scaled by 1/||q|| in LDS once, then all 32 A fragments
//     (64 VGPRs) hoisted into registers before the M loop -> the hot loop is
//     pure global_load_b64 (B from L2-resident kn) + v_wmma.
// ---------------------------------------------------------------------------

typedef __attribute__((ext_vector_type(2))) float v2f;
typedef __attribute__((ext_vector_type(8))) float v8f;

#define HH    16      // heads
#define DD    128     // head dim
#define TOPK  16      // neighbors kept
#define NCH   512     // memory keys processed per chunk
#define AST   132     // LDS stride for A tile (128+4 -> conflict-free)
#define SST   513     // LDS stride for sims tile (512+1 -> conflict-free scan)

// ---------------------------------------------------------------------------
// Kernel 1: row-wise L2 normalization (x / max(||x||, 1e-6)), D=128.
// One wave32 per row; each lane owns one float4.
// ---------------------------------------------------------------------------
__global__ __launch_bounds__(256) void l2norm_rows_kernel(
    const float* __restrict__ src, float* __restrict__ dst, int nrows) {
  const int row  = blockIdx.x * 8 + (threadIdx.x >> 5);
  const int lane = threadIdx.x & 31;
  if (row >= nrows) return;
  const float4 v = ((const float4*)(src + (size_t)row * DD))[lane];
  float ss = v.x * v.x + v.y * v.y + v.z * v.z + v.w * v.w;
#pragma unroll
  for (int off = 16; off > 0; off >>= 1) ss += __shfl_xor(ss, off, 32);
  const float inv = 1.0f / fmaxf(sqrtf(ss), 1e-6f);
  float4 o;
  o.x = v.x * inv; o.y = v.y * inv; o.z = v.z * inv; o.w = v.w * inv;
  ((float4*)(dst + (size_t)row * DD))[lane] = o;
}

// ---------------------------------------------------------------------------
// Kernel 2: fused sims-GEMM (WMMA f32 16x16x4) + top-16 + gather + gate.
// Grid: H * (S/16) blocks of 256 threads (8 waves).
// Each block: head h, queries [q0, q0+16).
// ---------------------------------------------------------------------------
__global__ __launch_bounds__(256) void praxis_mem_kernel(
    const float* __restrict__ query,    // [H,S,D]  (B=1)
    const float* __restrict__ outputs,  // [H,S,D]
    const float* __restrict__ gate,     // [H]
    const float* __restrict__ kn,       // [H,M,D]  pre-normalized keys (ws)
    const float* __restrict__ vmem,     // [H,M,D]  value memories
    float* __restrict__ out,            // [H,S,D]
    int S, int M) {
  __shared__ float Ald[16][AST];        // query tile (normalized in-place)
  __shared__ float qiv[16];             // per-row 1/max(||q||,eps)
  __shared__ float sims[16][SST];       // 16 x 512 sims chunk
  __shared__ float ksc[16][TOPK];       // top-k scores (sorted desc)
  __shared__ int   kid[16][TOPK];       // top-k indices

  const int tid  = threadIdx.x;
  const int lane = tid & 31;
  const int wave = tid >> 5;            // 0..7
  const int half = lane >> 4;           // 0 or 1 (lane half)
  const int ml   = lane & 15;           // M (A/C) or N (B) coordinate

  const int qtiles = S >> 4;
  const int h  = blockIdx.x / qtiles;
  const int q0 = (blockIdx.x % qtiles) << 4;

  const int arow = tid >> 4;            // 0..15 (row owned during staging)
  const int acol = (tid & 15) * 8;      // 8 floats per thread

  // ---- stage query tile into LDS (16 rows x 128, float4 vectorized) ----
  {
    const float4* s4 =
        (const float4*)(query + ((size_t)h * S + q0 + arow) * DD + acol);
    const float4 a = s4[0], b = s4[1];
    *(float4*)&Ald[arow][acol]     = a;
    *(float4*)&Ald[arow][acol + 4] = b;
  }
  __syncthreads();

  // ---- per-row inverse norms + top-k list init (threads 0..15) ----
  if (tid < 16) {
    float ss = 0.0f;
#pragma unroll 8
    for (int j = 0; j < DD; ++j) { const float x = Ald[tid][j]; ss += x * x; }
    qiv[tid] = 1.0f / fmaxf(sqrtf(ss), 1e-6f);
#pragma unroll
    for (int j = 0; j < TOPK; ++j) {
      ksc[tid][j] = -__builtin_inff();
      kid[tid][j] = 0;
    }
  }
  __syncthreads();

  // ---- normalize A tile in place (cooperative, 8 floats/thread) ----
  {
    const float s = qiv[arow];
#pragma unroll
    for (int j = 0; j < 8; ++j) Ald[arow][acol + j] *= s;
  }
  __syncthreads();

  // ---- hoist all 32 A fragments into registers (64 VGPRs) ----
  // A layout (16x4 f32): M = lane&15 on both halves; lanes 0-15 carry K=k0,
  // k0+1 and lanes 16-31 carry K=k0+2, k0+3 in the fragment pair.
  v2f afrag[DD / 4];
#pragma unroll
  for (int k0 = 0; k0 < DD; k0 += 4) {
    const int kb = k0 + half * 2;
    afrag[k0 >> 2].x = Ald[ml][kb];
    afrag[k0 >> 2].y = Ald[ml][kb + 1];
  }

  const float* knH = kn + (size_t)h * M * DD;
  float minS = -__builtin_inff();       // meaningful in threads 0..15

  for (int c0 = 0; c0 < M; c0 += NCH) {
    // prefetch next chunk of normalized keys toward L2
    if (c0 + NCH < M) {
      const float* p = knH + (size_t)(c0 + NCH) * DD;
      __builtin_prefetch(p + (size_t)tid * DD, 0, 1);
      __builtin_prefetch(p + (size_t)(tid + 256) * DD, 0, 1);
    }

    // ---- compute 16 x NCH sims tile with WMMA (all 8 waves, EXEC full) ----
#pragma unroll
    for (int t = 0; t < 4; ++t) {
      const int n0 = c0 + wave * 64 + t * 16;         // 16-col tile base
      v8f acc = {0.f, 0.f, 0.f, 0.f, 0.f, 0.f, 0.f, 0.f};
      // B fragment source row: kn[h][n0+ml][*], K pair = k0 + 2*half
      const float* brow = knH + (size_t)(n0 + ml) * DD + half * 2;
#pragma unroll
      for (int k0 = 0; k0 < DD; k0 += 4) {
        v2f b;
        b.x = brow[k0];                               // B: 4x16, N=ml
        b.y = brow[k0 + 1];
        acc = __builtin_amdgcn_wmma_f32_16x16x4_f32(
            /*neg_a=*/false, afrag[k0 >> 2], /*neg_b=*/false, b,
            /*c_mod=*/(short)0, acc, /*reuse_a=*/false, /*reuse_b=*/false);
      }
      // C layout: vgpr r -> row (r + 8*half), col = ml
#pragma unroll
      for (int r = 0; r < 8; ++r)
        sims[r + half * 8][(n0 - c0) + ml] = acc[r];
    }
    __syncthreads();

    // ---- per-query top-16 maintenance (thread q scans its sims row) ----
    if (tid < 16) {
      for (int j = 0; j < NCH; ++j) {
        const float v = sims[tid][j];
        if (v > minS) {
          int p = TOPK - 1;
          while (p > 0 && ksc[tid][p - 1] < v) {
            ksc[tid][p] = ksc[tid][p - 1];
            kid[tid][p] = kid[tid][p - 1];
            --p;
          }
          ksc[tid][p] = v;
          kid[tid][p] = c0 + j;
          minS = ksc[tid][TOPK - 1];
        }
      }
    }
    __syncthreads();
  }

  // ---- gather values, weighted sum, sigmoid-gate blend ----
  {
    const int q  = tid >> 4;            // query in tile
    const int d0 = (tid & 15) * 8;      // 8 D-elements per thread
    const float* Vh = vmem + (size_t)h * M * DD;
    float acc[8] = {0, 0, 0, 0, 0, 0, 0, 0};
#pragma unroll
    for (int j = 0; j < TOPK; ++j) {
      const float s  = ksc[q][j];
      const int  idx = kid[q][j];
      const float4* vp = (const float4*)(Vh + (size_t)idx * DD + d0);
      const float4 v0 = vp[0], v1 = vp[1];
      acc[0] += s * v0.x; acc[1] += s * v0.y;
      acc[2] += s * v0.z; acc[3] += s * v0.w;
      acc[4] += s * v1.x; acc[5] += s * v1.y;
      acc[6] += s * v1.z; acc[7] += s * v1.w;
    }
    const float g  = 1.0f / (1.0f + __expf(-gate[h]));
    const float og = 1.0f - g;
    const size_t obase = ((size_t)h * S + q0 + q) * (size_t)DD + d0;
    const float4* o4 = (const float4*)(outputs + obase);
    const float4 p0 = o4[0], p1 = o4[1];
    float4 r0, r1;
    r0.x = g * acc[0] + og * p0.x; r0.y = g * acc[1] + og * p0.y;
    r0.z = g * acc[2] + og * p0.z; r0.w = g * acc[3] + og * p0.w;
    r1.x = g * acc[4] + og * p1.x; r1.y = g * acc[5] + og * p1.y;
    r1.z = g * acc[6] + og * p1.z; r1.w = g * acc[7] + og * p1.w;
    *(float4*)(out + obase)     = r0;
    *(float4*)(out + obase + 4) = r1;
  }
}

// ---------------------------------------------------------------------------
// Host launcher.
// d_in order: inputs(0), query(1), key(2), value(3), outputs(4), gate(5),
//             key_memories(6), value_memories(7).  Output: [B,H,S,D] f32.
// Workspace: H*M*D floats (32 MB for reference dims) for normalized keys.
// ---------------------------------------------------------------------------
extern "C" void kernel_launch(void* const* d_in, const int* in_sizes, int n_in,
                              void* d_out, int out_size, void* d_ws,
                              size_t ws_size, hipStream_t stream) {
  const float* query   = (const float*)d_in[1];
  const float* outputs = (const float*)d_in[4];
  const float* gate    = (const float*)d_in[5];
  const float* key_mem = (const float*)d_in[6];
  const float* val_mem = (const float*)d_in[7];
  float*       out     = (float*)d_out;

  const int S = in_sizes[1] / (HH * DD);   // 2048
  const int M = in_sizes[6] / (HH * DD);   // 4096

  float* kn = (float*)d_ws;                // [H,M,D] normalized keys

  // 1) normalize key memories (one wave per row, 8 rows per block)
  const int krows = HH * M;
  l2norm_rows_kernel<<<(krows + 7) / 8, 256, 0, stream>>>(key_mem, kn, krows);

  // 2) fused sims + top-k + gather + gate
  const int blocks = HH * (S >> 4);
  praxis_mem_kernel<<<blocks, 256, 0, stream>>>(query, outputs, gate, kn,
                                                val_mem, out, S, M);
}